// WaveletTransformLayer_64355789963908
// MI455X (gfx1250) — compile-verified
//
#include <hip/hip_runtime.h>

typedef __attribute__((ext_vector_type(2))) float v2f;
typedef __attribute__((ext_vector_type(4))) float v4f;
typedef __attribute__((ext_vector_type(8))) float v8f;

#define ROW_LEN 16384
#define THREADS 256

// Coefficient of the fused 4-level Haar matrix H4 (16x16):
//  rows 0-7  : cA0_m = (x[2m] - x[2m+1]) * c
//  rows 8-11 : cA1_q = (x[4q]+x[4q+1]-x[4q+2]-x[4q+3]) * c^2
//  rows 12-13: cA2_q = (sum x[8q..8q+3] - sum x[8q+4..8q+7]) * c^3
//  row 14    : cA3   = (sum x[0..7] - sum x[8..15]) * c^4
//  row 15    : cD3   = (sum x[0..15]) * c^4   (carried to the next stage)
__device__ __forceinline__ float h4coef(int m, int k) {
  const float c  = 0.70710678118654752440f;
  const float c2 = c * c;
  const float c3 = c2 * c;
  const float c4 = c2 * c2;
  if (m < 8)  return (k == 2 * m) ? c : ((k == 2 * m + 1) ? -c : 0.0f);
  if (m < 12) { int q = m - 8;  return ((k >> 2) == q) ? (((k & 3) < 2) ? c2 : -c2) : 0.0f; }
  if (m < 14) { int q = m - 12; return ((k >> 3) == q) ? (((k & 7) < 4) ? c3 : -c3) : 0.0f; }
  if (m == 14) return (k < 8) ? c4 : -c4;
  return c4;
}

__global__ __launch_bounds__(THREADS) void haar_rows(const float* __restrict__ x,
                                                     float* __restrict__ out,
                                                     int nrows) {
  __shared__ float smem[ROW_LEN];                 // 64 KB: one full row
  const int row  = blockIdx.x;
  const int tid  = threadIdx.x;
  const int lane = tid & 31;
  const int wave = tid >> 5;
  const float c  = 0.70710678118654752440f;

  // ---------------- Stage A: async copy row (64 KB) global -> LDS -------------
  {
    const float*  rowp    = x + (size_t)row * ROW_LEN;
    const unsigned ldsBase = (unsigned)(size_t)&smem[0];
#pragma unroll
    for (int i = 0; i < 16; ++i) {
      const int      chunk = tid + THREADS * i;          // 4096 x 16B chunks
      const unsigned laddr = ldsBase + (unsigned)chunk * 16u;
      const float*   g     = rowp + chunk * 4;
      asm volatile("global_load_async_to_lds_b128 %0, %1, off"
                   :: "v"(laddr), "v"(g) : "memory");
    }
    asm volatile("s_wait_asynccnt 0" ::: "memory");      // own wave's asyncs done
  }
  __syncthreads();                                       // visible to all waves

  // ------- Stage B: levels 0..3 fused as D = H4 * X via v_wmma_f32_16x16x4_f32
  // A operand (16x4 chunk of H4): lane L, vgpr v -> M = L%16, K = 4j + v + 2*(L/16)
  const int mrow = lane & 15;
  const int hi   = lane >> 4;
  v2f aop[4];
#pragma unroll
  for (int j = 0; j < 4; ++j) {
    aop[j][0] = h4coef(mrow, 4 * j + 2 * hi);
    aop[j][1] = h4coef(mrow, 4 * j + 2 * hi + 1);
  }

  const size_t off1 = (size_t)nrows * 8192;              // cA level-1 block
  const size_t off2 = off1 + (size_t)nrows * 4096;       // cA level-2 block
  const size_t off3 = off2 + (size_t)nrows * 2048;       // cA level-3 block

  float cd3[8] = {};
#pragma unroll
  for (int t = 0; t < 8; ++t) {
    const int col = (wave * 8 + t) * 16 + mrow;          // 16-sample block index
    v8f d = {0.f, 0.f, 0.f, 0.f, 0.f, 0.f, 0.f, 0.f};
#pragma unroll
    for (int j = 0; j < 4; ++j) {
      // B operand (4x16): lane L, vgpr v -> K = 4j + v + 2*(L/16), N = L%16
      const v2f b = *(const v2f*)&smem[col * 16 + 4 * j + 2 * hi];
      d = __builtin_amdgcn_wmma_f32_16x16x4_f32(false, aop[j], false, b,
                                                (short)0, d, false, false);
    }
    // C/D layout: lane<16 holds rows 0..7, lane>=16 holds rows 8..15 of column `col`
    if (lane < 16) {
      float* p  = out + (size_t)row * 8192 + (size_t)col * 8;   // cA0[8*col .. +7]
      v4f lo = {d[0], d[1], d[2], d[3]};
      v4f hh = {d[4], d[5], d[6], d[7]};
      __builtin_nontemporal_store(lo, (v4f*)p);
      __builtin_nontemporal_store(hh, (v4f*)(p + 4));
    } else {
      v4f w1 = {d[0], d[1], d[2], d[3]};                        // cA1[4*col .. +3]
      __builtin_nontemporal_store(w1, (v4f*)(out + off1 + (size_t)row * 4096 + (size_t)col * 4));
      v2f w2 = {d[4], d[5]};                                    // cA2[2*col .. +1]
      __builtin_nontemporal_store(w2, (v2f*)(out + off2 + (size_t)row * 2048 + (size_t)col * 2));
      __builtin_nontemporal_store(d[6], out + off3 + (size_t)row * 1024 + col); // cA3[col]
      cd3[t] = d[7];                                            // cD3[col] -> stage C
    }
  }
  __syncthreads();                       // everyone done reading smem[0..16383]
  if (lane >= 16) {
#pragma unroll
    for (int t = 0; t < 8; ++t)
      smem[(wave * 8 + t) * 16 + mrow] = cd3[t];   // 1024-sample level-4 signal
  }
  __syncthreads();

  // ---------------- Stage C: levels 4..13 on smem[0..1023], in place ----------
  size_t off  = off3 + (size_t)nrows * 1024;       // start of level-4 cA block
  int    half = 512;
  while (half >= 1) {
    float e0 = 0.f, o0 = 0.f, e1 = 0.f, o1 = 0.f;
    const int  i0 = tid, i1 = tid + THREADS;
    const bool a0 = i0 < half, a1 = i1 < half;
    if (a0) { e0 = smem[2 * i0]; o0 = smem[2 * i0 + 1]; }
    if (a1) { e1 = smem[2 * i1]; o1 = smem[2 * i1 + 1]; }
    __syncthreads();                               // all reads before in-place writes
    if (a0) {
      smem[i0] = (e0 + o0) * c;                    // cD carried
      __builtin_nontemporal_store((e0 - o0) * c, out + off + (size_t)row * half + i0);
    }
    if (a1) {
      smem[i1] = (e1 + o1) * c;
      __builtin_nontemporal_store((e1 - o1) * c, out + off + (size_t)row * half + i1);
    }
    __syncthreads();
    off += (size_t)nrows * half;
    half >>= 1;
  }
  if (tid == 0)                                    // final length-1 cD
    out[off + row] = smem[0];
}

extern "C" void kernel_launch(void* const* d_in, const int* in_sizes, int n_in,
                              void* d_out, int out_size, void* d_ws, size_t ws_size,
                              hipStream_t stream) {
  (void)n_in; (void)out_size; (void)d_ws; (void)ws_size;
  const float* x   = (const float*)d_in[0];
  float*       out = (float*)d_out;
  const int nrows  = in_sizes[0] / ROW_LEN;        // 64*64 = 4096 rows
  haar_rows<<<nrows, THREADS, 0, stream>>>(x, out, nrows);
}